// graphsim_71768903516466
// MI455X (gfx1250) — compile-verified
//
#include <hip/hip_runtime.h>
#include <hip/hip_bf16.h>
#include <math.h>

// Problem constants (fixed by the reference)
#define NNODES 32768   // B*P
#define PNODES 512
#define NGRAPH 64
#define NEDGE  524288  // B*EG (per edge set, before self loops)
#define RSZ    30

typedef __attribute__((ext_vector_type(2))) float v2f;
typedef __attribute__((ext_vector_type(8))) float v8f;

// D = A(16x4,f32) * B(4x16,f32) + C(16x16,f32), full f32 precision WMMA.
__device__ __forceinline__ v8f wmma4(v2f a, v2f b, v8f c) {
  return __builtin_amdgcn_wmma_f32_16x16x4_f32(false, a, false, b, (short)0, c,
                                               false, false);
}

// ---------------- small utility kernels ----------------
__global__ void k_fill(float* __restrict__ p, float v, int n) {
  int i = blockIdx.x * blockDim.x + threadIdx.x;
  if (i < n) p[i] = v;
}

__global__ void k_deg(const int* __restrict__ dst, float* __restrict__ deg, int e) {
  int i = blockIdx.x * blockDim.x + threadIdx.x;
  if (i < e) atomicAdd(&deg[dst[i]], 1.0f);
}

__global__ void k_rsqrt(float* __restrict__ p, int n) {
  int i = blockIdx.x * blockDim.x + threadIdx.x;
  if (i < n) p[i] = rsqrtf(p[i]);   // deg >= 1 always (self loop)
}

__global__ void k_relu(float* __restrict__ p, int n) {
  int i = blockIdx.x * blockDim.x + threadIdx.x;
  if (i < n) p[i] = fmaxf(p[i], 0.0f);
}

// ---------------- WMMA GEMM: O[n,fout] = A[n,fin] @ W[fin,fout] ----------------
// One wave per 16x16 output tile; f32 16x16x4 WMMA, K stepped by 4.
__global__ __launch_bounds__(32) void k_gemm(const float* __restrict__ A,
                                             const float* __restrict__ W,
                                             float* __restrict__ O,
                                             int fin, int fout) {
  const int row0 = blockIdx.x * 16;
  const int col0 = blockIdx.y * 16;
  const int lane = threadIdx.x;
  const int m  = lane & 15;     // row (A) / col (B) within tile
  const int hi = lane >> 4;     // selects K pair
  v8f acc = {0.f, 0.f, 0.f, 0.f, 0.f, 0.f, 0.f, 0.f};
  const float* arow = A + (size_t)(row0 + m) * fin;
  for (int k0 = 0; k0 < fin; k0 += 4) {
    const int ka = k0 + 2 * hi;
    v2f a; a.x = arow[ka];                  a.y = arow[ka + 1];
    v2f b; b.x = W[ka * fout + col0 + m];   b.y = W[(ka + 1) * fout + col0 + m];
    acc = wmma4(a, b, acc);
  }
#pragma unroll
  for (int v = 0; v < 8; ++v)
    O[(size_t)(row0 + v + 8 * hi) * fout + col0 + m] = acc[v];
}

// out[v,f] = dinv[v]^2 * h[v,f] + bias[f]   (self-loop term + bias init)
__global__ void k_selfinit(const float* __restrict__ h, const float* __restrict__ dinv,
                           const float* __restrict__ bias, float* __restrict__ out,
                           int f) {
  int i = blockIdx.x * blockDim.x + threadIdx.x;
  if (i >= NNODES * f) return;
  int v = i / f, c = i - v * f;
  float dv = dinv[v];
  out[i] = dv * dv * h[i] + bias[c];
}

// out[dst] += dinv[src]*dinv[dst] * h[src]   (float4-vectorized atomics)
__global__ void k_edge(const int* __restrict__ src, const int* __restrict__ dst,
                       const float* __restrict__ dinv, const float* __restrict__ h,
                       float* __restrict__ out, int f) {
  int t = blockIdx.x * blockDim.x + threadIdx.x;
  const int g = f >> 2;
  if (t >= NEDGE * g) return;
  int e = t / g;
  int q = (t - e * g) << 2;
  int s = src[e], d = dst[e];
  float w = dinv[s] * dinv[d];
  float4 hv = *(const float4*)(h + (size_t)s * f + q);
  float* op = out + (size_t)d * f + q;
  atomicAdd(op + 0, w * hv.x);
  atomicAdd(op + 1, w * hv.y);
  atomicAdd(op + 2, w * hv.z);
  atomicAdd(op + 3, w * hv.w);
}

// ---------------- similarity + bilinear resize, one block per graph ----------------
// Only the 60 rows/cols sampled by the 30-out bilinear are needed: gather them
// (zero-padded to 64), run a 64x64xF WMMA GEMM in LDS, then emit the 30x30 map.
__global__ __launch_bounds__(512) void k_sim(const float* __restrict__ hs,
                                             const float* __restrict__ ht,
                                             float* __restrict__ smap, int f) {
  __shared__ float sS[64 * 68];
  __shared__ float sT[64 * 68];
  __shared__ float sM[64 * 64];
  const int b = blockIdx.x;
  const int tid = threadIdx.x;
  const float scale = (float)PNODES / (float)RSZ;

  for (int idx = tid; idx < 64 * f; idx += 512) {
    int r = idx / f, q = idx - r * f;
    float vs = 0.f, vt = 0.f;
    if (r < 60) {
      int i = (r < 30) ? r : r - 30;
      float c = (i + 0.5f) * scale - 0.5f;
      c = fminf(fmaxf(c, 0.0f), (float)(PNODES - 1));
      int c0 = (int)floorf(c);
      int node = (r < 30) ? c0 : min(c0 + 1, PNODES - 1);
      vs = hs[((size_t)b * PNODES + node) * f + q];
      vt = ht[((size_t)b * PNODES + node) * f + q];
    }
    sS[r * 68 + q] = vs;
    sT[r * 68 + q] = vt;
  }
  __syncthreads();

  // 16 waves -> 4x4 grid of 16x16 tiles. S = s @ t^T  (B frag == t rows).
  const int wave = tid >> 5, lane = tid & 31;
  const int tr = wave >> 2, tc = wave & 3;
  const int m = lane & 15, hi = lane >> 4;
  v8f acc = {0.f, 0.f, 0.f, 0.f, 0.f, 0.f, 0.f, 0.f};
  for (int k0 = 0; k0 < f; k0 += 4) {
    const int ka = k0 + 2 * hi;
    v2f a;  a.x  = sS[(tr * 16 + m) * 68 + ka];  a.y  = sS[(tr * 16 + m) * 68 + ka + 1];
    v2f bb; bb.x = sT[(tc * 16 + m) * 68 + ka];  bb.y = sT[(tc * 16 + m) * 68 + ka + 1];
    acc = wmma4(a, bb, acc);
  }
#pragma unroll
  for (int v = 0; v < 8; ++v)
    sM[(tr * 16 + v + 8 * hi) * 64 + tc * 16 + m] = acc[v];
  __syncthreads();

  // bilinear: rows 0..29 = y0, rows 30..59 = y1 (same for cols)
  for (int o = tid; o < RSZ * RSZ; o += 512) {
    int i = o / RSZ, j = o - i * RSZ;
    float cy = fminf(fmaxf((i + 0.5f) * scale - 0.5f, 0.f), (float)(PNODES - 1));
    float cx = fminf(fmaxf((j + 0.5f) * scale - 0.5f, 0.f), (float)(PNODES - 1));
    float wy = cy - floorf(cy);
    float wx = cx - floorf(cx);
    float s00 = sM[i * 64 + j];
    float s01 = sM[i * 64 + j + 30];
    float s10 = sM[(i + 30) * 64 + j];
    float s11 = sM[(i + 30) * 64 + j + 30];
    smap[(size_t)b * (RSZ * RSZ) + o] =
        (1.f - wy) * ((1.f - wx) * s00 + wx * s01) +
        wy        * ((1.f - wx) * s10 + wx * s11);
  }
}

// ---------------- CNN branch: one block per (graph, branch), all in LDS ----------------
__global__ __launch_bounds__(256) void k_cnn(const float* __restrict__ smaps,
    const float* __restrict__ w1, const float* __restrict__ b1,
    const float* __restrict__ w2, const float* __restrict__ b2,
    const float* __restrict__ w3, const float* __restrict__ b3,
    const float* __restrict__ w4, const float* __restrict__ b4,
    float* __restrict__ feats) {
  __shared__ float bufA[6400];   // input(900) / conv2 out(16*400) / ch sums(32)
  __shared__ float bufB[8192];   // conv1 out(8*625) / conv3 out(32*256)
  const int b = blockIdx.x;
  const int br = blockIdx.y;
  const int tid = threadIdx.x;

  const float* inmap = smaps + ((size_t)br * NGRAPH + b) * (RSZ * RSZ);
  for (int i = tid; i < 900; i += 256) bufA[i] = inmap[i];
  __syncthreads();

  // conv1: 1->8, 6x6, 30->25, relu
  {
    const float* W = w1 + br * 8 * 36;
    const float* Bb = b1 + br * 8;
    for (int o = tid; o < 8 * 625; o += 256) {
      int ch = o / 625, p = o - ch * 625, y = p / 25, x = p - y * 25;
      float acc = Bb[ch];
      const float* wp = W + ch * 36;
      for (int ky = 0; ky < 6; ++ky)
        for (int kx = 0; kx < 6; ++kx)
          acc += bufA[(y + ky) * 30 + x + kx] * wp[ky * 6 + kx];
      bufB[ch * 625 + p] = fmaxf(acc, 0.f);
    }
  }
  __syncthreads();
  // conv2: 8->16, 6x6, 25->20, relu
  {
    const float* W = w2 + br * 16 * 8 * 36;
    const float* Bb = b2 + br * 16;
    for (int o = tid; o < 16 * 400; o += 256) {
      int ch = o / 400, p = o - ch * 400, y = p / 20, x = p - y * 20;
      float acc = Bb[ch];
      for (int c = 0; c < 8; ++c) {
        const float* wp = W + (ch * 8 + c) * 36;
        const float* ip = bufB + c * 625;
        for (int ky = 0; ky < 6; ++ky)
          for (int kx = 0; kx < 6; ++kx)
            acc += ip[(y + ky) * 25 + x + kx] * wp[ky * 6 + kx];
      }
      bufA[ch * 400 + p] = fmaxf(acc, 0.f);
    }
  }
  __syncthreads();
  // conv3: 16->32, 5x5, 20->16, relu
  {
    const float* W = w3 + br * 32 * 16 * 25;
    const float* Bb = b3 + br * 32;
    for (int o = tid; o < 32 * 256; o += 256) {
      int ch = o / 256, p = o - ch * 256, y = p >> 4, x = p & 15;
      float acc = Bb[ch];
      for (int c = 0; c < 16; ++c) {
        const float* wp = W + (ch * 16 + c) * 25;
        const float* ip = bufA + c * 400;
        for (int ky = 0; ky < 5; ++ky)
          for (int kx = 0; kx < 5; ++kx)
            acc += ip[(y + ky) * 20 + x + kx] * wp[ky * 5 + kx];
      }
      bufB[ch * 256 + p] = fmaxf(acc, 0.f);
    }
  }
  __syncthreads();
  if (tid < 32) bufA[tid] = 0.f;   // per-channel sum accumulators
  __syncthreads();
  // conv4: 32->32, 5x5, 16->12 (no relu), accumulate sums for global mean
  {
    const float* W = w4 + br * 32 * 32 * 25;
    const float* Bb = b4 + br * 32;
    for (int o = tid; o < 32 * 144; o += 256) {
      int ch = o / 144, p = o - ch * 144, y = p / 12, x = p - y * 12;
      float acc = Bb[ch];
      for (int c = 0; c < 32; ++c) {
        const float* wp = W + (ch * 32 + c) * 25;
        const float* ip = bufB + c * 256;
        for (int ky = 0; ky < 5; ++ky)
          for (int kx = 0; kx < 5; ++kx)
            acc += ip[(y + ky) * 16 + x + kx] * wp[ky * 5 + kx];
      }
      atomicAdd(&bufA[ch], acc);
    }
  }
  __syncthreads();
  if (tid < 32)
    feats[(size_t)b * 96 + br * 32 + tid] = bufA[tid] * (1.0f / 144.0f);
}

// ---------------- MLP head: one thread per graph ----------------
__global__ __launch_bounds__(64) void k_mlp(const float* __restrict__ feats,
    const float* __restrict__ w1, const float* __restrict__ b1,
    const float* __restrict__ w2, const float* __restrict__ b2,
    const float* __restrict__ w3, const float* __restrict__ b3,
    float* __restrict__ out) {
  int g = threadIdx.x;
  if (g >= NGRAPH) return;
  const float* fr = feats + g * 96;
  float h1[16], h2[16];
  for (int o = 0; o < 16; ++o) {
    float a = b1[o];
    for (int c = 0; c < 96; ++c) a += fr[c] * w1[c * 16 + o];
    h1[o] = fmaxf(a, 0.f);
  }
  for (int o = 0; o < 16; ++o) {
    float a = b2[o];
    for (int c = 0; c < 16; ++c) a += h1[c] * w2[c * 16 + o];
    h2[o] = fmaxf(a, 0.f);
  }
  float a = b3[0];
  for (int c = 0; c < 16; ++c) a += h2[c] * w3[c];
  out[g] = 1.0f / (1.0f + expf(-a));
}

extern "C" void kernel_launch(void* const* d_in, const int* in_sizes, int n_in,
                              void* d_out, int out_size, void* d_ws, size_t ws_size,
                              hipStream_t stream) {
  const float* x_i = (const float*)d_in[0];
  const float* x_j = (const float*)d_in[1];
  const int* ei = (const int*)d_in[2];
  const int* ej = (const int*)d_in[3];
  const float* gw1 = (const float*)d_in[5];  const float* gb1 = (const float*)d_in[6];
  const float* gw2 = (const float*)d_in[7];  const float* gb2 = (const float*)d_in[8];
  const float* gw3 = (const float*)d_in[9];  const float* gb3 = (const float*)d_in[10];
  const float* cw1 = (const float*)d_in[11]; const float* cb1 = (const float*)d_in[12];
  const float* cw2 = (const float*)d_in[13]; const float* cb2 = (const float*)d_in[14];
  const float* cw3 = (const float*)d_in[15]; const float* cb3 = (const float*)d_in[16];
  const float* cw4 = (const float*)d_in[17]; const float* cb4 = (const float*)d_in[18];
  const float* lw1 = (const float*)d_in[19]; const float* lb1 = (const float*)d_in[20];
  const float* lw2 = (const float*)d_in[21]; const float* lb2 = (const float*)d_in[22];
  const float* lw3 = (const float*)d_in[23]; const float* lb3 = (const float*)d_in[24];

  float* ws = (float*)d_ws;
  const size_t N = NNODES;
  float* dinv[2] = { ws, ws + N };
  float* hbuf = ws + 2 * N;                       // 64N scratch (pre-agg GEMM out)
  float* h1[2] = { ws + 66 * N, ws + 178 * N };   // 64N each
  float* h2[2] = { ws + 130 * N, ws + 242 * N };  // 32N each
  float* h3[2] = { ws + 162 * N, ws + 274 * N };  // 16N each
  float* smaps = ws + 290 * N;                    // 3*64*900
  float* feats = smaps + 3 * NGRAPH * RSZ * RSZ;  // 64*96

  for (int s = 0; s < 2; ++s) {
    const float* x = s ? x_j : x_i;
    const int* esrc = s ? ej : ei;
    const int* edst = esrc + NEDGE;
    // degrees (self loop => init 1.0) -> dinv
    k_fill<<<NNODES / 256, 256, 0, stream>>>(dinv[s], 1.0f, NNODES);
    k_deg<<<NEDGE / 256, 256, 0, stream>>>(edst, dinv[s], NEDGE);
    k_rsqrt<<<NNODES / 256, 256, 0, stream>>>(dinv[s], NNODES);
    // layer 1: 64 -> 64
    k_gemm<<<dim3(NNODES / 16, 4), 32, 0, stream>>>(x, gw1, hbuf, 64, 64);
    k_selfinit<<<(NNODES * 64) / 256, 256, 0, stream>>>(hbuf, dinv[s], gb1, h1[s], 64);
    k_edge<<<(NEDGE * 16) / 256, 256, 0, stream>>>(esrc, edst, dinv[s], hbuf, h1[s], 64);
    k_relu<<<(NNODES * 64) / 256, 256, 0, stream>>>(h1[s], NNODES * 64);
    // layer 2: 64 -> 32
    k_gemm<<<dim3(NNODES / 16, 2), 32, 0, stream>>>(h1[s], gw2, hbuf, 64, 32);
    k_selfinit<<<(NNODES * 32) / 256, 256, 0, stream>>>(hbuf, dinv[s], gb2, h2[s], 32);
    k_edge<<<(NEDGE * 8) / 256, 256, 0, stream>>>(esrc, edst, dinv[s], hbuf, h2[s], 32);
    k_relu<<<(NNODES * 32) / 256, 256, 0, stream>>>(h2[s], NNODES * 32);
    // layer 3: 32 -> 16
    k_gemm<<<dim3(NNODES / 16, 1), 32, 0, stream>>>(h2[s], gw3, hbuf, 32, 16);
    k_selfinit<<<(NNODES * 16) / 256, 256, 0, stream>>>(hbuf, dinv[s], gb3, h3[s], 16);
    k_edge<<<(NEDGE * 4) / 256, 256, 0, stream>>>(esrc, edst, dinv[s], hbuf, h3[s], 16);
    k_relu<<<(NNODES * 16) / 256, 256, 0, stream>>>(h3[s], NNODES * 16);
  }

  // similarity maps (only the 60 bilinear-sampled rows/cols are computed)
  k_sim<<<NGRAPH, 512, 0, stream>>>(h1[0], h1[1], smaps, 64);
  k_sim<<<NGRAPH, 512, 0, stream>>>(h2[0], h2[1], smaps + NGRAPH * RSZ * RSZ, 32);
  k_sim<<<NGRAPH, 512, 0, stream>>>(h3[0], h3[1], smaps + 2 * NGRAPH * RSZ * RSZ, 16);

  // CNN branches + head
  k_cnn<<<dim3(NGRAPH, 3), 256, 0, stream>>>(smaps, cw1, cb1, cw2, cb2,
                                             cw3, cb3, cw4, cb4, feats);
  k_mlp<<<1, 64, 0, stream>>>(feats, lw1, lb1, lw2, lb2, lw3, lb3, (float*)d_out);
}